// ContextQueryAttentionLayer_79611513798797
// MI455X (gfx1250) — compile-verified
//
#include <hip/hip_runtime.h>

// ContextQueryAttention (BiDAF) for MI455X / gfx1250, fp32 WMMA (16x16x4).
// Inputs: context (64,1024,128) f32, query (64,128,128) f32,
//         c_mask (64,1024) bool(u8), q_mask (64,128) bool(u8), w (384,) f32.
// Output: (64,1024,512) f32 = concat(ctx, c2q, ctx*c2q, ctx*q2c).
// Workspace: S = B*N*M f32 (33.5MB), T = B*M*D f32 (4.2MB).

#define NEG_INF -1000000000.0f

typedef __attribute__((ext_vector_type(2))) float v2f;
typedef __attribute__((ext_vector_type(8))) float v8f;

constexpr int CB = 64;    // batch
constexpr int CN = 1024;  // context length
constexpr int CM = 128;   // query length
constexpr int CD = 128;   // feature dim

__device__ __forceinline__ v8f v8zero() {
  v8f z = {0.f, 0.f, 0.f, 0.f, 0.f, 0.f, 0.f, 0.f};
  return z;
}

// D = A(16x4) * B(4x16) + C, all fp32.
__device__ __forceinline__ v8f wmma4(v2f a, v2f b, v8f c) {
  return __builtin_amdgcn_wmma_f32_16x16x4_f32(
      false, a, false, b, (short)0, c, false, false);
}

// ---------------------------------------------------------------------------
// Kernel 1: S[b,i,j] = qs[j] + cs[i] + dot(ctx[b,i]*w_m, q[b,j])
// grid (N/128, B), block 256 (8 waves; wave w owns 16 i-rows x all 128 j).
// Dynamic LDS ~138KB.
// ---------------------------------------------------------------------------
__global__ __launch_bounds__(256)
void k1_scores(const float* __restrict__ ctx, const float* __restrict__ qry,
               const float* __restrict__ w, float* __restrict__ S)
{
  extern __shared__ float sm[];
  float* q_lds = sm;                    // 128*132
  float* c_lds = q_lds + 128 * 132;     // 128*132 (raw ctx, then ctx*w_m)
  float* w_lds = c_lds + 128 * 132;     // 384
  float* qs    = w_lds + 384;           // 128
  float* cs    = qs + 128;              // 128

  const int b    = blockIdx.y;
  const int i0   = blockIdx.x * 128;
  const int tid  = threadIdx.x;
  const int lane = tid & 31;
  const int wv   = tid >> 5;

  for (int idx = tid; idx < 128 * 128; idx += 256) {
    int r = idx >> 7, d = idx & 127;
    q_lds[r * 132 + d] = qry[(b * CM + r) * CD + d];
    c_lds[r * 132 + d] = ctx[(b * CN + i0 + r) * CD + d];
  }
  for (int idx = tid; idx < 384; idx += 256) w_lds[idx] = w[idx];
  __syncthreads();

  // qs[j] = q[j].w_q (threads 0..127); cs[i] = ctx[i].w_c (threads 128..255)
  if (tid < 128) {
    float s = 0.f;
    for (int d = 0; d < 128; ++d) s += q_lds[tid * 132 + d] * w_lds[d];
    qs[tid] = s;
  } else {
    int r = tid - 128;
    float s = 0.f;
    for (int d = 0; d < 128; ++d) s += c_lds[r * 132 + d] * w_lds[128 + d];
    cs[r] = s;
  }
  __syncthreads();

  // pre-scale context tile by w_m (A operand becomes plain LDS reads)
  for (int idx = tid; idx < 128 * 128; idx += 256) {
    int r = idx >> 7, d = idx & 127;
    c_lds[r * 132 + d] *= w_lds[256 + d];
  }
  __syncthreads();

  const int arow = wv * 16 + (lane & 15);
  const int koff = (lane >> 4) << 1;
  const int jloc = lane & 15;
  const int aoff = arow * 132;

  v8f acc[8];
#pragma unroll
  for (int t = 0; t < 8; ++t) acc[t] = v8zero();

  v2f a_cur, b_cur[8];
  {
    int k = koff;
    a_cur.x = c_lds[aoff + k];
    a_cur.y = c_lds[aoff + k + 1];
#pragma unroll
    for (int t = 0; t < 8; ++t) {
      int j = t * 16 + jloc;
      b_cur[t].x = q_lds[j * 132 + k];
      b_cur[t].y = q_lds[j * 132 + k + 1];
    }
  }

  for (int k0 = 0; k0 < 124; k0 += 4) {
    const int k = k0 + 4 + koff;
    v2f a_nxt, b_nxt[8];
    a_nxt.x = c_lds[aoff + k];
    a_nxt.y = c_lds[aoff + k + 1];
#pragma unroll
    for (int t = 0; t < 8; ++t) {
      int j = t * 16 + jloc;
      b_nxt[t].x = q_lds[j * 132 + k];
      b_nxt[t].y = q_lds[j * 132 + k + 1];
    }
#pragma unroll
    for (int t = 0; t < 8; ++t) acc[t] = wmma4(a_cur, b_cur[t], acc[t]);
    a_cur = a_nxt;
#pragma unroll
    for (int t = 0; t < 8; ++t) b_cur[t] = b_nxt[t];
  }
#pragma unroll
  for (int t = 0; t < 8; ++t) acc[t] = wmma4(a_cur, b_cur[t], acc[t]);

  const int rb = wv * 16 + ((lane >> 4) << 3);
#pragma unroll
  for (int t = 0; t < 8; ++t) {
    int j = t * 16 + jloc;
#pragma unroll
    for (int v = 0; v < 8; ++v) {
      int r = rb + v;
      S[(b * CN + i0 + r) * CM + j] = acc[t][v] + cs[r] + qs[j];
    }
  }
}

// ---------------------------------------------------------------------------
// Kernel 2: column softmax (over i, c_mask) and T = Bm^T * ctx.
// grid (M/16, B), block 256 (8 waves, one 16x16 d-tile each).
// ---------------------------------------------------------------------------
__global__ __launch_bounds__(256)
void k2_colsm_t(const float* __restrict__ ctx,
                const unsigned char* __restrict__ cmask,
                const float* __restrict__ S, float* __restrict__ T)
{
  __shared__ float bm[64 * 17];     // Bm chunk [i_local][j_local], padded
  __shared__ float cbias[1024];
  __shared__ float red_m[256];
  __shared__ float red_s[256];
  __shared__ float colmax[16];
  __shared__ float colrcp[16];

  const int b    = blockIdx.y;
  const int j0   = blockIdx.x * 16;
  const int tid  = threadIdx.x;
  const int lane = tid & 31;
  const int wv   = tid >> 5;

  for (int idx = tid; idx < 1024; idx += 256)
    cbias[idx] = cmask[b * CN + idx] ? 0.f : NEG_INF;
  __syncthreads();

  // Phase A: online max/sum over i per column (16 segments of 64).
  {
    const int c = tid & 15, seg = tid >> 4;
    float m = -3.4e38f, s = 0.f;
    for (int i = seg * 64; i < seg * 64 + 64; ++i) {
      float v  = S[(b * CN + i) * CM + j0 + c] + cbias[i];
      float nm = fmaxf(m, v);
      s = s * __expf(m - nm) + __expf(v - nm);
      m = nm;
    }
    red_m[seg * 16 + c] = m;
    red_s[seg * 16 + c] = s;
  }
  __syncthreads();
  if (tid < 16) {
    float m = -3.4e38f;
    for (int seg = 0; seg < 16; ++seg) m = fmaxf(m, red_m[seg * 16 + tid]);
    float s = 0.f;
    for (int seg = 0; seg < 16; ++seg)
      s += red_s[seg * 16 + tid] * __expf(red_m[seg * 16 + tid] - m);
    colmax[tid] = m;
    colrcp[tid] = 1.f / s;
  }
  __syncthreads();

  // Phase B: T strip = Bm^T * ctx over 16 chunks of 64 rows, pipelined.
  const int dtile = wv * 16;
  const int jloc  = lane & 15;
  const int koff  = (lane >> 4) << 1;
  v8f acc = v8zero();

  for (int ic = 0; ic < 16; ++ic) {
    const int ibase = ic * 64;
    for (int idx = tid; idx < 64 * 16; idx += 256) {
      int r = idx >> 4, c = idx & 15;
      float v = S[(b * CN + ibase + r) * CM + j0 + c];
      bm[r * 17 + c] = __expf(v + cbias[ibase + r] - colmax[c]) * colrcp[c];
    }
    __syncthreads();

    const float* cbase = ctx + (b * CN + ibase) * CD + dtile + jloc;
    v2f a_cur, b_cur;
    {
      int k = koff;
      a_cur.x = bm[k * 17 + jloc];
      a_cur.y = bm[(k + 1) * 17 + jloc];
      b_cur.x = cbase[k * CD];
      b_cur.y = cbase[(k + 1) * CD];
    }
    for (int kk0 = 0; kk0 < 60; kk0 += 4) {
      const int k = kk0 + 4 + koff;
      v2f a_nxt, b_nxt;
      a_nxt.x = bm[k * 17 + jloc];
      a_nxt.y = bm[(k + 1) * 17 + jloc];
      b_nxt.x = cbase[k * CD];
      b_nxt.y = cbase[(k + 1) * CD];
      acc = wmma4(a_cur, b_cur, acc);
      a_cur = a_nxt;
      b_cur = b_nxt;
    }
    acc = wmma4(a_cur, b_cur, acc);
    __syncthreads();
  }

  const int jl8 = (lane >> 4) << 3;
#pragma unroll
  for (int v = 0; v < 8; ++v)
    T[(b * CM + j0 + jl8 + v) * CD + dtile + jloc] = acc[v];
}

// ---------------------------------------------------------------------------
// Kernel 3: row softmax A (over j, q_mask); c2q = A*q; q2c = A*T; write out.
// grid (N/64, B), block 256. Dynamic LDS ~203KB (q and T staged in LDS).
// ---------------------------------------------------------------------------
__global__ __launch_bounds__(256)
void k3_out(const float* __restrict__ ctx, const float* __restrict__ qry,
            const unsigned char* __restrict__ qmask,
            const float* __restrict__ S, const float* __restrict__ T,
            float* __restrict__ out)
{
  extern __shared__ float sm[];
  float* af    = sm;                   // 64*132: S tile -> A -> later c2q
  float* gf    = af + 64 * 132;        // 64*132: q2c
  float* q_lds = gf + 64 * 132;        // 128*132
  float* t_lds = q_lds + 128 * 132;    // 128*132
  float* qb    = t_lds + 128 * 132;    // 128

  const int b    = blockIdx.y;
  const int i0   = blockIdx.x * 64;
  const int tid  = threadIdx.x;
  const int lane = tid & 31;
  const int wv   = tid >> 5;

  if (tid < 128) qb[tid] = qmask[b * CM + tid] ? 0.f : NEG_INF;

  for (int idx = tid; idx < 128 * 128; idx += 256) {
    int r = idx >> 7, d = idx & 127;
    q_lds[r * 132 + d] = qry[(b * CM + r) * CD + d];
    t_lds[r * 132 + d] = T[(b * CM + r) * CD + d];
  }
  __syncthreads();

  for (int idx = tid; idx < 64 * 128; idx += 256) {
    int r = idx >> 7, j = idx & 127;
    af[r * 132 + j] = S[(b * CN + i0 + r) * CM + j] + qb[j];
  }
  __syncthreads();

  // row softmax: 4 lanes per row, 32 j's each, shuffle-reduce within quad
  {
    const int r = tid >> 2, c4 = tid & 3;
    float m = -3.4e38f;
    for (int j = c4 * 32; j < c4 * 32 + 32; ++j)
      m = fmaxf(m, af[r * 132 + j]);
    m = fmaxf(m, __shfl_xor(m, 1, 32));
    m = fmaxf(m, __shfl_xor(m, 2, 32));
    float s = 0.f;
    for (int j = c4 * 32; j < c4 * 32 + 32; ++j) {
      float e = __expf(af[r * 132 + j] - m);
      af[r * 132 + j] = e;
      s += e;
    }
    s += __shfl_xor(s, 1, 32);
    s += __shfl_xor(s, 2, 32);
    float rs = 1.f / s;
    for (int j = c4 * 32; j < c4 * 32 + 32; ++j) af[r * 132 + j] *= rs;
  }
  __syncthreads();

  // waves 0-3: c2q = A * q_lds; waves 4-7: q2c = A * t_lds
  const int itile = wv & 3;
  const float* src = (wv < 4) ? q_lds : t_lds;
  const int arow = itile * 16 + (lane & 15);
  const int koff = (lane >> 4) << 1;
  const int nloc = lane & 15;
  const int aoff = arow * 132;

  v8f acc[8];
#pragma unroll
  for (int t = 0; t < 8; ++t) acc[t] = v8zero();

  v2f a_cur, b_cur[8];
  {
    int k = koff;
    a_cur.x = af[aoff + k];
    a_cur.y = af[aoff + k + 1];
#pragma unroll
    for (int dt = 0; dt < 8; ++dt) {
      b_cur[dt].x = src[k * 132 + dt * 16 + nloc];
      b_cur[dt].y = src[(k + 1) * 132 + dt * 16 + nloc];
    }
  }

  for (int k0 = 0; k0 < 124; k0 += 4) {
    const int k = k0 + 4 + koff;
    v2f a_nxt, b_nxt[8];
    a_nxt.x = af[aoff + k];
    a_nxt.y = af[aoff + k + 1];
#pragma unroll
    for (int dt = 0; dt < 8; ++dt) {
      b_nxt[dt].x = src[k * 132 + dt * 16 + nloc];
      b_nxt[dt].y = src[(k + 1) * 132 + dt * 16 + nloc];
    }
#pragma unroll
    for (int dt = 0; dt < 8; ++dt) acc[dt] = wmma4(a_cur, b_cur[dt], acc[dt]);
    a_cur = a_nxt;
#pragma unroll
    for (int dt = 0; dt < 8; ++dt) b_cur[dt] = b_nxt[dt];
  }
#pragma unroll
  for (int dt = 0; dt < 8; ++dt) acc[dt] = wmma4(a_cur, b_cur[dt], acc[dt]);

  __syncthreads();  // everyone done reading af

  float* dstl = (wv < 4) ? af : gf;
  const int rb = itile * 16 + ((lane >> 4) << 3);
#pragma unroll
  for (int dt = 0; dt < 8; ++dt) {
#pragma unroll
    for (int v = 0; v < 8; ++v)
      dstl[(rb + v) * 132 + dt * 16 + nloc] = acc[dt][v];
  }
  __syncthreads();

  for (int idx = tid; idx < 64 * 128; idx += 256) {
    int r = idx >> 7, d = idx & 127;
    float cv  = ctx[(b * CN + i0 + r) * CD + d];
    float c2q = af[r * 132 + d];
    float q2c = gf[r * 132 + d];
    float* o = out + (size_t)(b * CN + i0 + r) * (4 * CD) + d;
    o[0]      = cv;
    o[CD]     = c2q;
    o[2 * CD] = cv * c2q;
    o[3 * CD] = cv * q2c;
  }
}

// ---------------------------------------------------------------------------
extern "C" void kernel_launch(void* const* d_in, const int* in_sizes, int n_in,
                              void* d_out, int out_size, void* d_ws, size_t ws_size,
                              hipStream_t stream)
{
  const float* ctx          = (const float*)d_in[0];
  const float* qry          = (const float*)d_in[1];
  const unsigned char* cmsk = (const unsigned char*)d_in[2];  // numpy bool = u8
  const unsigned char* qmsk = (const unsigned char*)d_in[3];
  const float* w            = (const float*)d_in[4];
  float* out                = (float*)d_out;

  float* S = (float*)d_ws;                     // B*N*M f32
  float* T = S + (size_t)CB * CN * CM;         // B*M*D f32

  size_t sh1 = (size_t)(2 * 128 * 132 + 384 + 256) * sizeof(float);
  k1_scores<<<dim3(CN / 128, CB), 256, sh1, stream>>>(ctx, qry, w, S);

  k2_colsm_t<<<dim3(CM / 16, CB), 256, 0, stream>>>(ctx, cmsk, S, T);

  size_t sh3 = (size_t)(2 * 64 * 132 + 2 * 128 * 132 + 128) * sizeof(float);
  k3_out<<<dim3(CN / 64, CB), 256, sh3, stream>>>(ctx, qry, qmsk, S, T, out);
}